// InitialPoseModel_6760278524532
// MI455X (gfx1250) — compile-verified
//
#include <hip/hip_runtime.h>
#include <stdint.h>

// InitialPoseModel fused kernel for MI455X (gfx1250, wave32).
// One workgroup per batch: top-10 selection -> candidate gather -> std-filter
// vote -> WMMA cross-covariance -> 3x3 SVD Kabsch.

#define BATCH   64
#define NPTS    12288
#define NKP     9
#define KCAND   10
#define THREADS 256
#define VEC4_PER_THREAD (NPTS / (THREADS * 4))   // 12 float4 loads per thread

typedef float v2f __attribute__((ext_vector_type(2)));
typedef float v8f __attribute__((ext_vector_type(8)));

__device__ __forceinline__ void topk_insert(unsigned long long* top, unsigned long long key) {
    if (key > top[KCAND - 1]) {
        int p = KCAND - 1;
        while (p > 0 && top[p - 1] < key) { top[p] = top[p - 1]; --p; }
        top[p] = key;
    }
}

__global__ __launch_bounds__(THREADS) void pose_kernel(
    const float* __restrict__ pcld,   // [b, n, 3]
    const float* __restrict__ kofs,   // [b, n, 8, 3]
    const float* __restrict__ cofs,   // [b, n, 1, 3]
    const float* __restrict__ seg,    // [b, n, 1]
    const float* __restrict__ mesh,   // [b, 9, 3]
    float* __restrict__ out)          // R[64*9] | t[64*3] | voted[64*27]
{
    const int b   = blockIdx.x;
    const int tid = threadIdx.x;

    __shared__ unsigned long long sCand[THREADS * KCAND];  // 20 KB tournament arena
    __shared__ int   sIdx[KCAND];
    __shared__ float sKC[NKP][KCAND][3];
    __shared__ float sVoted[NKP][3];
    __shared__ float sCA[3], sCB[3];
    // WMMA operand staging, zero-padded to the full 16-lane frame so wave 0
    // issues unconditional ds_loads (no per-lane EXEC save/restore).
    __shared__ float sAmP[12][16];   // A[M][K] stored as [K][M]: Am^T, 3x9 valid
    __shared__ float sBmP[12][16];   // B[K][N]: Bm, 9x3 valid
    __shared__ float sH[3][3];

    // ---------- Phase 1: per-thread top-10 over a strided float4 chunk ----------
    unsigned long long top[KCAND];
    #pragma unroll
    for (int i = 0; i < KCAND; ++i) top[i] = 0ull;

    const float4* seg4 = (const float4*)(seg + (size_t)b * NPTS);
    for (int i = 0; i < VEC4_PER_THREAD; ++i) {
        int v = tid + i * THREADS;
        float4 s4 = seg4[v];                     // coalesced 16B loads
        int base = v * 4;
        // key = score_bits (positive floats: bit order == value order) | ~idx
        // (larger key wins; on tie, smaller index wins — matches lax.top_k)
        topk_insert(top, ((unsigned long long)__float_as_uint(s4.x) << 32) | (unsigned)(~(base + 0)));
        topk_insert(top, ((unsigned long long)__float_as_uint(s4.y) << 32) | (unsigned)(~(base + 1)));
        topk_insert(top, ((unsigned long long)__float_as_uint(s4.z) << 32) | (unsigned)(~(base + 2)));
        topk_insert(top, ((unsigned long long)__float_as_uint(s4.w) << 32) | (unsigned)(~(base + 3)));
    }
    #pragma unroll
    for (int i = 0; i < KCAND; ++i) sCand[tid * KCAND + i] = top[i];
    __syncthreads();

    // Tournament merge of 256 sorted 10-lists (8 levels).
    for (int stride = 1; stride < THREADS; stride <<= 1) {
        if ((tid & (2 * stride - 1)) == 0) {
            unsigned long long mo[KCAND];
            const unsigned long long* A = &sCand[(size_t)tid * KCAND];
            const unsigned long long* B = &sCand[(size_t)(tid + stride) * KCAND];
            int ia = 0, ib = 0;
            #pragma unroll
            for (int i = 0; i < KCAND; ++i) mo[i] = (A[ia] >= B[ib]) ? A[ia++] : B[ib++];
            #pragma unroll
            for (int i = 0; i < KCAND; ++i) sCand[(size_t)tid * KCAND + i] = mo[i];
        }
        __syncthreads();
    }
    if (tid < KCAND) sIdx[tid] = (int)(~(unsigned)(sCand[tid] & 0xffffffffull));
    __syncthreads();

    // ---------- Phase 2: gather 9x10 candidate keypoint positions ----------
    if (tid < NKP * KCAND) {
        int kpt = tid / KCAND, c = tid % KCAND;
        int i = sIdx[c];
        size_t pbase = ((size_t)b * NPTS + i) * 3;
        const float* ofs = (kpt < 8) ? (kofs + (((size_t)b * NPTS + i) * 8 + kpt) * 3)
                                     : (cofs + pbase);
        #pragma unroll
        for (int d = 0; d < 3; ++d)
            sKC[kpt][c][d] = pcld[pbase + d] + ofs[d];
    }
    __syncthreads();

    // ---------- Phase 3: std-filtered mean vote per keypoint ----------
    if (tid < NKP) {
        float m[3] = {0, 0, 0}, var[3] = {0, 0, 0};
        for (int c = 0; c < KCAND; ++c)
            for (int d = 0; d < 3; ++d) m[d] += sKC[tid][c][d];
        for (int d = 0; d < 3; ++d) m[d] *= (1.0f / KCAND);
        for (int c = 0; c < KCAND; ++c)
            for (int d = 0; d < 3; ++d) { float e = sKC[tid][c][d] - m[d]; var[d] += e * e; }
        float sd[3];
        for (int d = 0; d < 3; ++d) sd[d] = sqrtf(var[d] * (1.0f / KCAND));  // population std
        float wsum = 0.0f, acc[3] = {0, 0, 0};
        for (int c = 0; c < KCAND; ++c) {
            bool in = true;
            for (int d = 0; d < 3; ++d) in = in && (fabsf(sKC[tid][c][d] - m[d]) <= sd[d]);
            float w = in ? 1.0f : 0.0f;
            wsum += w;
            for (int d = 0; d < 3; ++d) acc[d] += sKC[tid][c][d] * w;
        }
        float inv = 1.0f / (wsum + 1e-6f);
        for (int d = 0; d < 3; ++d) {
            float vv = acc[d] * inv;
            sVoted[tid][d] = vv;
            out[576 + 192 + ((size_t)b * NKP + tid) * 3 + d] = vv;   // kpts_voted output
        }
    }
    __syncthreads();

    // ---------- Phase 4: centroids + centered, zero-padded WMMA operands ----------
    if (tid < 3) {
        float a = 0.0f, bb = 0.0f;
        for (int n = 0; n < NKP; ++n) {
            a  += mesh[((size_t)b * NKP + n) * 3 + tid];
            bb += sVoted[n][tid];
        }
        sCA[tid] = a * (1.0f / NKP);
        sCB[tid] = bb * (1.0f / NKP);
    }
    __syncthreads();
    if (tid < 12 * 16) {                 // fill full padded frames (zeros outside 9x3)
        int n = tid / 16, col = tid % 16;
        bool valid = (n < NKP) && (col < 3);
        sAmP[n][col] = valid ? mesh[((size_t)b * NKP + n) * 3 + col] - sCA[col] : 0.0f;
        sBmP[n][col] = valid ? sVoted[n][col] - sCB[col] : 0.0f;
    }
    __syncthreads();

    // ---------- Phase 5: H = Am^T * Bm via chained V_WMMA_F32_16X16X4_F32 ----------
    // A = Am^T (3x12 in a 16xK frame), B = Bm (12x3 in a Kx16 frame), K chunks of 4.
    // Wave 0 only; uniform branch so inactive waves jump over (EXEC all-ones inside).
    if (tid < 32) {
        int half = tid >> 4;     // lanes 16-31 hold K = k+2, k+3
        int lane = tid & 15;
        v8f acc = {0, 0, 0, 0, 0, 0, 0, 0};
        #pragma unroll
        for (int c = 0; c < 3; ++c) {
            int kk = c * 4 + half * 2;
            v2f a, bm;
            a.x  = sAmP[kk    ][lane];   // A[M=lane][K=kk]   (unconditional ds_load)
            a.y  = sAmP[kk + 1][lane];
            bm.x = sBmP[kk    ][lane];   // B[K=kk][N=lane]
            bm.y = sBmP[kk + 1][lane];
            acc = __builtin_amdgcn_wmma_f32_16x16x4_f32(
                false, a, false, bm, (short)0, acc, false, false);
        }
        // D layout: VGPR r -> M=r, N=lane (lanes 0-15). H[i][j] = acc[i] on lane j.
        if (half == 0 && lane < 3) {
            sH[0][lane] = acc[0];
            sH[1][lane] = acc[1];
            sH[2][lane] = acc[2];
        }
    }
    __syncthreads();

    // ---------- Phase 6: 3x3 SVD Kabsch (one thread per batch) ----------
    if (tid == 0) {
        float H[3][3];
        for (int i = 0; i < 3; ++i)
            for (int j = 0; j < 3; ++j) H[i][j] = sH[i][j];

        // S = H^T H (symmetric), Jacobi eigendecomposition -> V
        float S[3][3];
        for (int i = 0; i < 3; ++i)
            for (int j = 0; j < 3; ++j) {
                float s = 0.0f;
                for (int k = 0; k < 3; ++k) s += H[k][i] * H[k][j];
                S[i][j] = s;
            }
        float V[3][3] = {{1, 0, 0}, {0, 1, 0}, {0, 0, 1}};
        for (int sweep = 0; sweep < 6; ++sweep) {
            for (int pp = 0; pp < 3; ++pp) {
                int p = (pp == 2) ? 1 : 0;
                int q = (pp == 0) ? 1 : 2;
                float apq = S[p][q];
                if (fabsf(apq) < 1e-20f) continue;
                float app = S[p][p], aqq = S[q][q];
                float tau = (aqq - app) / (2.0f * apq);
                float tt = (tau >= 0.0f) ? 1.0f / (tau + sqrtf(1.0f + tau * tau))
                                         : 1.0f / (tau - sqrtf(1.0f + tau * tau));
                float c = 1.0f / sqrtf(1.0f + tt * tt), s = tt * c;
                S[p][p] = app - tt * apq;
                S[q][q] = aqq + tt * apq;
                S[p][q] = S[q][p] = 0.0f;
                int r = 3 - p - q;
                float arp = S[r][p], arq = S[r][q];
                S[r][p] = S[p][r] = c * arp - s * arq;
                S[r][q] = S[q][r] = s * arp + c * arq;
                for (int a2 = 0; a2 < 3; ++a2) {
                    float vp = V[a2][p], vq = V[a2][q];
                    V[a2][p] = c * vp - s * vq;
                    V[a2][q] = s * vp + c * vq;
                }
            }
        }
        // Sort eigenpairs descending (SVD convention: smallest sigma last)
        float lam[3] = {S[0][0], S[1][1], S[2][2]};
        int ord[3] = {0, 1, 2};
        for (int i = 0; i < 2; ++i)
            for (int j = i + 1; j < 3; ++j)
                if (lam[ord[j]] > lam[ord[i]]) { int tm = ord[i]; ord[i] = ord[j]; ord[j] = tm; }
        float v0[3], v1[3], v2c[3];
        for (int a2 = 0; a2 < 3; ++a2) {
            v0[a2]  = V[a2][ord[0]];
            v1[a2]  = V[a2][ord[1]];
            v2c[a2] = V[a2][ord[2]];
        }
        // Left singular vectors: u_i = normalize(H v_i); u2 = u0 x u1
        float w0[3], w1[3];
        for (int a2 = 0; a2 < 3; ++a2) {
            w0[a2] = H[a2][0] * v0[0] + H[a2][1] * v0[1] + H[a2][2] * v0[2];
            w1[a2] = H[a2][0] * v1[0] + H[a2][1] * v1[1] + H[a2][2] * v1[2];
        }
        float n0 = sqrtf(w0[0] * w0[0] + w0[1] * w0[1] + w0[2] * w0[2]);
        float i0 = 1.0f / fmaxf(n0, 1e-12f);
        float u0[3] = {w0[0] * i0, w0[1] * i0, w0[2] * i0};
        float d01 = w1[0] * u0[0] + w1[1] * u0[1] + w1[2] * u0[2];
        for (int a2 = 0; a2 < 3; ++a2) w1[a2] -= d01 * u0[a2];
        float n1 = sqrtf(w1[0] * w1[0] + w1[1] * w1[1] + w1[2] * w1[2]);
        float i1 = 1.0f / fmaxf(n1, 1e-12f);
        float u1[3] = {w1[0] * i1, w1[1] * i1, w1[2] * i1};
        float u2[3] = {u0[1] * u1[2] - u0[2] * u1[1],
                       u0[2] * u1[0] - u0[0] * u1[2],
                       u0[0] * u1[1] - u0[1] * u1[0]};
        // det(U)=+1 by construction; reflection fix s = sign(det V)
        float cx = v1[1] * v2c[2] - v1[2] * v2c[1];
        float cy = v1[2] * v2c[0] - v1[0] * v2c[2];
        float cz = v1[0] * v2c[1] - v1[1] * v2c[0];
        float detV = v0[0] * cx + v0[1] * cy + v0[2] * cz;
        float sf = (detV < 0.0f) ? -1.0f : 1.0f;
        // R = v0 u0^T + v1 u1^T + s * v2 u2^T  (== V diag(1,1,s) U^T)
        float R[3][3];
        for (int a2 = 0; a2 < 3; ++a2)
            for (int c2 = 0; c2 < 3; ++c2)
                R[a2][c2] = v0[a2] * u0[c2] + v1[a2] * u1[c2] + sf * v2c[a2] * u2[c2];
        for (int a2 = 0; a2 < 3; ++a2)
            for (int c2 = 0; c2 < 3; ++c2)
                out[(size_t)b * 9 + a2 * 3 + c2] = R[a2][c2];
        for (int a2 = 0; a2 < 3; ++a2)
            out[576 + (size_t)b * 3 + a2] =
                sCB[a2] - (R[a2][0] * sCA[0] + R[a2][1] * sCA[1] + R[a2][2] * sCA[2]);
    }
}

extern "C" void kernel_launch(void* const* d_in, const int* in_sizes, int n_in,
                              void* d_out, int out_size, void* d_ws, size_t ws_size,
                              hipStream_t stream) {
    const float* pcld = (const float*)d_in[0];   // [64,12288,3]
    const float* kofs = (const float*)d_in[1];   // [64,12288,8,3]
    const float* cofs = (const float*)d_in[2];   // [64,12288,1,3]
    const float* seg  = (const float*)d_in[3];   // [64,12288,1]
    const float* mesh = (const float*)d_in[4];   // [64,9,3]
    pose_kernel<<<BATCH, THREADS, 0, stream>>>(pcld, kofs, cofs, seg, mesh, (float*)d_out);
}